// MultiViewRenderer_48180943126589
// MI455X (gfx1250) — compile-verified
//
#include <hip/hip_runtime.h>
#include <math.h>

// ---------------------------------------------------------------------------
// Soft multi-view depth rasterizer for gfx1250 (MI455X).
// WMMA f32 16x16x4 computes logw = -d2/sigma^2 + (1-depth)/gamma for a
// 16-pixel x 16-point tile per instruction; VALU does exp + masked accumulate.
// ---------------------------------------------------------------------------

typedef __attribute__((ext_vector_type(2))) float v2f;
typedef __attribute__((ext_vector_type(8))) float v8f;

#define IMG_S        64
#define NPTS         512
#define INV_SIG2     4096.0f      // 1/sigma^2, sigma = 1/64
#define INV_GAMMA    0.2f         // 1/gamma, gamma = 5
#define TANHF        0.57735026919f  // tan(30 deg)
#define INV_TANHF    1.73205080757f  // 1/tan(30 deg)
#define ZNEARC       0.1f
#define ZFARC        6.0f
#define BGW          1e-8f
#define PI_F         3.14159265358979f

__device__ __forceinline__ void norm3(float& x, float& y, float& z) {
    float n = sqrtf(x * x + y * y + z * z);
    float inv = 1.0f / fmaxf(n, 1e-12f);
    x *= inv; y *= inv; z *= inv;
}
__device__ __forceinline__ void cross3(float ax, float ay, float az,
                                       float bx, float by, float bz,
                                       float& cx, float& cy, float& cz) {
    cx = ay * bz - az * by;
    cy = az * bx - ax * bz;
    cz = ax * by - ay * bx;
}

__global__ __launch_bounds__(256)
void MultiViewRenderer_48180943126589_kernel(
    const float* __restrict__ vertices,   // (512,3)
    const float* __restrict__ radii,      // (512,)
    const float* __restrict__ at_centers, // (8,3)
    const float* __restrict__ at_normals, // (8,3)
    const float* __restrict__ rho,        // (8,)
    const float* __restrict__ phi,        // (8,)
    const float* __restrict__ theta,      // (8,)
    const float* __restrict__ up,         // (3,)
    float* __restrict__ out)              // (8,8,1,64,64)
{
    // per-point record: {bx, by, c, depth, thr, pad, pad, pad}  (32B stride)
    __shared__ float rec[NPTS * 8];

    const int view = blockIdx.y;        // 0..63
    const int b = view >> 3;
    const int n = view & 7;

    // ---------------- build MVP (redundant per thread) ----------------------
    // angles are clamped to [0, 2pi]: hardware sin/cos approximations are fine
    float rho_c = fminf(fmaxf(rho[n], 0.3f), 1.0f);
    float phi_c = fminf(fmaxf(phi[n], 0.0f), 0.5f * PI_F);
    float th_c  = fminf(fmaxf(theta[n], 0.0f), 2.0f * PI_F);
    float sp = __sinf(phi_c), cp = __cosf(phi_c);
    float st = __sinf(th_c),  ct = __cosf(th_c);
    float exl = rho_c * sp * ct, eyl = rho_c * sp * st, ezl = rho_c * cp;

    float cxc = at_centers[3 * b + 0], cyc = at_centers[3 * b + 1], czc = at_centers[3 * b + 2];
    float azx = at_normals[3 * b + 0], azy = at_normals[3 * b + 1], azz = at_normals[3 * b + 2];
    norm3(azx, azy, azz);
    float unx = up[0], uny = up[1], unz = up[2];
    norm3(unx, uny, unz);                       // normalized up (reused)
    float axx, axy, axz;
    cross3(unx, uny, unz, azx, azy, azz, axx, axy, axz);
    norm3(axx, axy, axz);
    float ayx, ayy, ayz;
    cross3(azx, azy, azz, axx, axy, axz, ayx, ayy, ayz);
    norm3(ayx, ayy, ayz);
    // eye = exl*ax + eyl*ay + ezl*az + center
    float ex = exl * axx + eyl * ayx + ezl * azx + cxc;
    float ey = exl * axy + eyl * ayy + ezl * azy + cyc;
    float ez = exl * axz + eyl * ayz + ezl * azz + czc;
    // front = normalize(eye - center)
    float fx = ex - cxc, fy = ey - cyc, fz = ez - czc;
    norm3(fx, fy, fz);
    // right = normalize(cross(up_n, front)); updir = normalize(cross(front, right))
    float rx, ry, rz;
    cross3(unx, uny, unz, fx, fy, fz, rx, ry, rz);
    norm3(rx, ry, rz);
    float ux, uy, uz;
    cross3(fx, fy, fz, rx, ry, rz, ux, uy, uz);
    norm3(ux, uy, uz);
    // translation
    float t0 = -(rx * ex + ry * ey + rz * ez);
    float t1 = -(ux * ex + uy * ey + uz * ez);
    float t2 = -(fx * ex + fy * ey + fz * ez);
    // mvp rows (row2 / NDC-z never used by the output)
    float r0x = INV_TANHF * rx, r0y = INV_TANHF * ry, r0z = INV_TANHF * rz, r0w = INV_TANHF * t0;
    float r1x = INV_TANHF * ux, r1y = INV_TANHF * uy, r1z = INV_TANHF * uz, r1w = INV_TANHF * t1;
    float r3x = -fx, r3y = -fy, r3z = -fz, r3w = -t2;   // w_clip = camera depth

    // ---------------- project all points into LDS ---------------------------
    for (int pt = threadIdx.x; pt < NPTS; pt += 256) {
        float vx = vertices[3 * pt + 0], vy = vertices[3 * pt + 1], vz = vertices[3 * pt + 2];
        float c0 = r0x * vx + r0y * vy + r0z * vz + r0w;
        float c1 = r1x * vx + r1y * vy + r1z * vz + r1w;
        float wc = r3x * vx + r3y * vy + r3z * vz + r3w;
        float wcs = (wc > 1e-6f) ? wc : 1.0f;
        float xs = c0 / wcs;
        float ys = c1 / wcs;
        float depth = fminf(fmaxf((wc - ZNEARC) * (1.0f / (ZFARC - ZNEARC)), 0.0f), 1.0f);
        bool  infr = (wc > ZNEARC) && (wc < ZFARC);
        float rn = radii[pt] / (wcs * TANHF);
        float cut2 = 25.0f * rn * rn;
        float cj = -(xs * xs + ys * ys) * INV_SIG2 + (1.0f - depth) * INV_GAMMA;
        float thr = infr ? ((1.0f - depth) * INV_GAMMA - cut2 * INV_SIG2) : 3.0e38f;
        float* r8 = &rec[pt * 8];
        r8[0] = 2.0f * xs * INV_SIG2;
        r8[1] = 2.0f * ys * INV_SIG2;
        r8[2] = cj;
        r8[3] = depth;
        r8[4] = thr;
    }
    __syncthreads();

    // ---------------- WMMA rasterization ------------------------------------
    const int wave = threadIdx.x >> 5;
    const int lane = threadIdx.x & 31;
    const int hi   = lane >> 4;          // 0: K=0,1   1: K=2,3
    const int colL = lane & 15;          // point column within chunk
    const int pixbase = blockIdx.x * 128 + wave * 16;   // 16 pixels per wave

    // A row_m = [px, py, 1, 0]
    v2f A;
    {
        int p = pixbase + (lane & 15);
        float pxv = ((float)(p & 63) + 0.5f) * (2.0f / 64.0f) - 1.0f;
        float pyv = 1.0f - ((float)(p >> 6) + 0.5f) * (2.0f / 64.0f);
        A.x = hi ? 1.0f : pxv;
        A.y = hi ? 0.0f : pyv;
    }
    // C[m,j] = -(px^2+py^2)/sigma^2  (column-independent)
    v8f C;
#pragma unroll
    for (int r = 0; r < 8; ++r) {
        int p = pixbase + r + hi * 8;
        float pxv = ((float)(p & 63) + 0.5f) * (2.0f / 64.0f) - 1.0f;
        float pyv = 1.0f - ((float)(p >> 6) + 0.5f) * (2.0f / 64.0f);
        C[r] = -(pxv * pxv + pyv * pyv) * INV_SIG2;
    }

    float accw[8], accd[8];
#pragma unroll
    for (int r = 0; r < 8; ++r) { accw[r] = 0.0f; accd[r] = 0.0f; }

#pragma unroll 4
    for (int k = 0; k < NPTS / 16; ++k) {
        int j = k * 16 + colL;
        const float4 q = *reinterpret_cast<const float4*>(&rec[j * 8]); // bx,by,c,depth
        const float thr = rec[j * 8 + 4];
        v2f B;
        B.x = hi ? q.z : q.x;            // K=0:2x/s2, K=2:c
        B.y = hi ? 0.0f : q.y;           // K=1:2y/s2, K=3:0
        v8f D = __builtin_amdgcn_wmma_f32_16x16x4_f32(
            /*neg_a=*/false, A, /*neg_b=*/false, B,
            /*c_mod=*/(short)0, C, /*reuse_a=*/false, /*reuse_b=*/false);
#pragma unroll
        for (int r = 0; r < 8; ++r) {
            float lw = D[r];
            float w = (lw >= thr) ? __expf(lw) : 0.0f;   // cutoff + frustum mask
            accw[r] += w;
            accd[r] += w * q.w;
        }
    }

    // reduce across the 16 point-columns (stays inside each 16-lane half)
#pragma unroll
    for (int r = 0; r < 8; ++r) {
        float w = accw[r], d = accd[r];
        w += __shfl_xor(w, 8, 32);  d += __shfl_xor(d, 8, 32);
        w += __shfl_xor(w, 4, 32);  d += __shfl_xor(d, 4, 32);
        w += __shfl_xor(w, 2, 32);  d += __shfl_xor(d, 2, 32);
        w += __shfl_xor(w, 1, 32);  d += __shfl_xor(d, 1, 32);
        accw[r] = w; accd[r] = d;
    }

    if (colL == 0) {
        int base = view * (IMG_S * IMG_S) + pixbase + hi * 8;
#pragma unroll
        for (int r = 0; r < 8; ++r)
            out[base + r] = (BGW + accd[r]) / (BGW + accw[r]);
    }
}

extern "C" void kernel_launch(void* const* d_in, const int* in_sizes, int n_in,
                              void* d_out, int out_size, void* d_ws, size_t ws_size,
                              hipStream_t stream) {
    const float* vertices   = (const float*)d_in[0];
    const float* radii      = (const float*)d_in[1];
    const float* at_centers = (const float*)d_in[2];
    const float* at_normals = (const float*)d_in[3];
    const float* rho        = (const float*)d_in[4];
    const float* phi        = (const float*)d_in[5];
    const float* theta      = (const float*)d_in[6];
    const float* up         = (const float*)d_in[7];

    dim3 grid(32, 64);   // 32 tiles of 128 pixels  x  64 (batch,view) cameras
    dim3 block(256);     // 8 wave32s; each wave owns a 16-pixel WMMA row tile
    MultiViewRenderer_48180943126589_kernel<<<grid, block, 0, stream>>>(
        vertices, radii, at_centers, at_normals, rho, phi, theta, up,
        (float*)d_out);
}